// YoloNMS_11647951307533
// MI455X (gfx1250) — compile-verified
//
#include <hip/hip_runtime.h>
#include <stdint.h>

#define N_ANCHORS 100800
#define DIMS      117
#define NCLS      80
#define NMASK     32
#define CONF_T    0.25f
#define IOU_T     0.45f
#define MAX_DET   300
#define TOPK      4096
#define NSORT     131072           // next pow2 >= N_ANCHORS
#define KWORDS    (TOPK / 32)      // 128 suppression words per row
#define JT        256              // j-tile per block in sup kernel
#define ITILE     16               // i rows per block in sup kernel
#define SORT_BLK  512              // threads per local-sort block
#define CHUNK     1024             // elements per local-sort block

typedef __attribute__((ext_vector_type(2))) float v2f;
typedef __attribute__((ext_vector_type(8))) float v8f;

// ---- helpers -------------------------------------------------------------

// Map float to a u32 whose ASCENDING order equals DESCENDING float order.
static __device__ __forceinline__ unsigned sortable_desc(float f) {
  unsigned u = __float_as_uint(f);
  u = (u & 0x80000000u) ? ~u : (u | 0x80000000u); // ascending-sortable
  return ~u;                                      // flip -> descending
}

static __device__ __forceinline__ unsigned ballot32(bool p) {
#if __has_builtin(__builtin_amdgcn_ballot_w32)
  return __builtin_amdgcn_ballot_w32(p);
#else
  return (unsigned)__ballot(p);
#endif
}

// CDNA5 async memory->LDS copies (gfx1250). VDST = LDS byte offset (low 32
// bits of the generic shared pointer), VADDR = 64-bit global address, GV mode.
static __device__ __forceinline__ void async_copy_b32(void* lds_dst, const void* gsrc) {
  unsigned dst = (unsigned)(unsigned long long)lds_dst;
  unsigned long long src = (unsigned long long)gsrc;
  asm volatile("global_load_async_to_lds_b32 %0, %1, off"
               :: "v"(dst), "v"(src) : "memory");
}
static __device__ __forceinline__ void async_copy_b64(void* lds_dst, const void* gsrc) {
  unsigned dst = (unsigned)(unsigned long long)lds_dst;
  unsigned long long src = (unsigned long long)gsrc;
  asm volatile("global_load_async_to_lds_b64 %0, %1, off"
               :: "v"(dst), "v"(src) : "memory");
}
static __device__ __forceinline__ void wait_async0() {
  asm volatile("s_wait_asynccnt 0" ::: "memory");
}

// ---- kernel 1: decode scores / classes / boxes, build sort keys ----------

__global__ void prep_kernel(const float* __restrict__ x,
                            float* __restrict__ scores_m,
                            float* __restrict__ classes,
                            float* __restrict__ yxyx,
                            unsigned long long* __restrict__ keys) {
  int i = blockIdx.x * blockDim.x + threadIdx.x;
  if (i >= NSORT) return;
  if (i < N_ANCHORS) {
    const float* row = x + (long long)i * DIMS;
    float xc = row[0], yc = row[1], w = row[2], h = row[3], obj = row[4];
    float best = -1.0f;  // products are >= 0, so first iter always sets best
    int bc = 0;
    #pragma unroll 8
    for (int c = 0; c < NCLS; ++c) {
      float pv = row[5 + c] * obj;  // argmax on products, first-max tie rule
      if (pv > best) { best = pv; bc = c; }
    }
    float sm = (obj > CONF_T) ? best : -1.0f;
    scores_m[i] = sm;
    classes[i]  = (float)bc;
    float hy = h * 0.5f, hx = w * 0.5f;
    yxyx[i * 4 + 0] = yc - hy;
    yxyx[i * 4 + 1] = xc - hx;
    yxyx[i * 4 + 2] = yc + hy;
    yxyx[i * 4 + 3] = xc + hx;
    keys[i] = ((unsigned long long)sortable_desc(sm) << 32) | (unsigned)i;
  } else {
    // padding sorts after every real entry (score -2 < -1)
    keys[i] = ((unsigned long long)sortable_desc(-2.0f) << 32) | (unsigned)i;
  }
}

// ---- kernel 2a: fully sort each 1024-chunk in LDS (k = 2..1024) ----------

__global__ void __launch_bounds__(SORT_BLK) bitonic_local_full(unsigned long long* __restrict__ keys) {
  __shared__ unsigned long long sk[CHUNK];
  int t = threadIdx.x;
  int base = blockIdx.x * CHUNK;
  async_copy_b64(&sk[t], keys + base + t);
  async_copy_b64(&sk[t + SORT_BLK], keys + base + t + SORT_BLK);
  wait_async0();
  __syncthreads();
  for (int kk = 2; kk <= CHUNK; kk <<= 1) {
    for (int jj = kk >> 1; jj > 0; jj >>= 1) {
      int i = ((t / jj) * (jj << 1)) + (t % jj);
      int ixj = i + jj;
      bool up = (((base + i) & kk) == 0);
      unsigned long long a = sk[i], b = sk[ixj];
      if ((a > b) == up) { sk[i] = b; sk[ixj] = a; }
      __syncthreads();
    }
  }
  keys[base + t] = sk[t];
  keys[base + t + SORT_BLK] = sk[t + SORT_BLK];
}

// ---- kernel 2b: global bitonic step (j >= 1024) --------------------------

__global__ void bitonic_step(unsigned long long* __restrict__ keys, int j, int k) {
  int i = blockIdx.x * blockDim.x + threadIdx.x;
  int ixj = i ^ j;
  if (ixj > i) {
    unsigned long long a = keys[i], b = keys[ixj];
    bool up = ((i & k) == 0);
    if ((a > b) == up) { keys[i] = b; keys[ixj] = a; }
  }
}

// ---- kernel 2c: finish merge phase in LDS (j = 512..1 for given k) -------

__global__ void __launch_bounds__(SORT_BLK) bitonic_local_merge(unsigned long long* __restrict__ keys, int kk) {
  __shared__ unsigned long long sk[CHUNK];
  int t = threadIdx.x;
  int base = blockIdx.x * CHUNK;
  async_copy_b64(&sk[t], keys + base + t);
  async_copy_b64(&sk[t + SORT_BLK], keys + base + t + SORT_BLK);
  wait_async0();
  __syncthreads();
  for (int jj = SORT_BLK; jj > 0; jj >>= 1) {
    int i = ((t / jj) * (jj << 1)) + (t % jj);
    int ixj = i + jj;
    bool up = (((base + i) & kk) == 0);
    unsigned long long a = sk[i], b = sk[ixj];
    if ((a > b) == up) { sk[i] = b; sk[ixj] = a; }
    __syncthreads();
  }
  keys[base + t] = sk[t];
  keys[base + t + SORT_BLK] = sk[t + SORT_BLK];
}

// ---- kernel 3: extract top-4096 (indices, scores, SoA boxes) -------------

__global__ void extract_kernel(const unsigned long long* __restrict__ keys,
                               const float* __restrict__ scores_m,
                               const float* __restrict__ yxyx,
                               float* __restrict__ top_s, int* __restrict__ top_i,
                               float* __restrict__ by1, float* __restrict__ bx1,
                               float* __restrict__ by2, float* __restrict__ bx2) {
  int t = blockIdx.x * blockDim.x + threadIdx.x;
  if (t >= TOPK) return;
  unsigned long long key = keys[t];
  int idx = (int)(unsigned)(key & 0xffffffffull);
  top_i[t] = idx;
  top_s[t] = scores_m[idx];
  by1[t] = yxyx[idx * 4 + 0];
  bx1[t] = yxyx[idx * 4 + 1];
  by2[t] = yxyx[idx * 4 + 2];
  bx2[t] = yxyx[idx * 4 + 3];
}

// ---- kernel 4: suppression bit-matrix  sup[i][j] = (iou>T) && (j>i) ------
// Pairwise union base (area_i + area_j) computed on the matrix unit with
// V_WMMA_F32_16X16X4_F32 (A = [area_i, 1, 0, 0], B = [1, area_j, 0, 0]^T);
// VALU does intersection min/max; wave32 ballot packs suppression bits.
// Block: 256 threads = 8 waves, tile = 16 i-rows x 256 j-cols.

__global__ void __launch_bounds__(JT) sup_kernel(const float* __restrict__ by1,
                                                 const float* __restrict__ bx1,
                                                 const float* __restrict__ by2,
                                                 const float* __restrict__ bx2,
                                                 unsigned* __restrict__ sup) {
  __shared__ float sjy1[JT], sjx1[JT], sjy2[JT], sjx2[JT];   // j tile boxes
  __shared__ float siy1[ITILE], six1[ITILE], siy2[ITILE], six2[ITILE]; // i tile

  int tid  = threadIdx.x;
  int lane = tid & 31;
  int wv   = tid >> 5;
  int j0   = blockIdx.x * JT;       // global j base of block tile
  int i0   = blockIdx.y * ITILE;    // global i base of block tile

  async_copy_b32(&sjy1[tid], by1 + j0 + tid);
  async_copy_b32(&sjx1[tid], bx1 + j0 + tid);
  async_copy_b32(&sjy2[tid], by2 + j0 + tid);
  async_copy_b32(&sjx2[tid], bx2 + j0 + tid);
  if (tid < ITILE) {
    async_copy_b32(&siy1[tid], by1 + i0 + tid);
    async_copy_b32(&six1[tid], bx1 + i0 + tid);
    async_copy_b32(&siy2[tid], by2 + i0 + tid);
    async_copy_b32(&six2[tid], bx2 + i0 + tid);
  }
  wait_async0();
  __syncthreads();

  bool lo16 = lane < 16;
  int  nl   = lane & 15;            // column within 16-wide WMMA tile
  int  wj   = wv * 32;              // wave's j offset within block tile

  int c0 = wj + nl;                 // tile0 column for this lane
  int c1 = c0 + 16;                 // tile1 column for this lane
  float jy1a = sjy1[c0], jx1a = sjx1[c0], jy2a = sjy2[c0], jx2a = sjx2[c0];
  float jy1b = sjy1[c1], jx1b = sjx1[c1], jy2b = sjy2[c1], jx2b = sjx2[c1];

  // Union base: area_i + area_j via WMMA outer-sum (or VALU fallback).
  float ub0[8], ub1[8];
#if __has_builtin(__builtin_amdgcn_wmma_f32_16x16x4_f32)
  {
    float ai = 0.0f, aj0 = 0.0f, aj1 = 0.0f;
    if (lo16) {
      ai  = (siy2[lane] - siy1[lane]) * (six2[lane] - six1[lane]);
      int q0 = wj + lane, q1 = q0 + 16;
      aj0 = (sjy2[q0] - sjy1[q0]) * (sjx2[q0] - sjx1[q0]);
      aj1 = (sjy2[q1] - sjy1[q1]) * (sjx2[q1] - sjx1[q1]);
    }
    v2f Aop; Aop.x = ai;                 Aop.y = lo16 ? 1.0f : 0.0f;
    v2f B0;  B0.x  = lo16 ? 1.0f : 0.0f; B0.y  = aj0;
    v2f B1;  B1.x  = lo16 ? 1.0f : 0.0f; B1.y  = aj1;
    v8f Z = {};
    v8f D0 = __builtin_amdgcn_wmma_f32_16x16x4_f32(false, Aop, false, B0,
                                                   (short)0, Z, false, false);
    v8f D1 = __builtin_amdgcn_wmma_f32_16x16x4_f32(false, Aop, false, B1,
                                                   (short)0, Z, false, false);
    #pragma unroll
    for (int r = 0; r < 8; ++r) { ub0[r] = D0[r]; ub1[r] = D1[r]; }
  }
#else
  {
    float aja = (jy2a - jy1a) * (jx2a - jx1a);
    float ajb = (jy2b - jy1b) * (jx2b - jx1b);
    int mrow = lo16 ? 0 : 8;
    #pragma unroll
    for (int r = 0; r < 8; ++r) {
      int m = mrow + r;
      float aim = (siy2[m] - siy1[m]) * (six2[m] - six1[m]);
      ub0[r] = aim + aja;
      ub1[r] = aim + ajb;
    }
  }
#endif

  int mrow = lo16 ? 0 : 8;
  int jg0  = j0 + c0;
  int jg1  = j0 + c1;
  unsigned bal0[8], bal1[8];
  #pragma unroll
  for (int r = 0; r < 8; ++r) {
    int m  = mrow + r;
    int ig = i0 + m;
    float iy1 = siy1[m], ix1 = six1[m], iy2 = siy2[m], ix2 = six2[m];

    float ih0 = fminf(iy2, jy2a) - fmaxf(iy1, jy1a);
    float iw0 = fminf(ix2, jx2a) - fmaxf(ix1, jx1a);
    float in0 = fmaxf(ih0, 0.0f) * fmaxf(iw0, 0.0f);
    float un0 = ub0[r] - in0;
    bool  p0  = (in0 / fmaxf(un0, 1e-9f) > IOU_T) && (jg0 > ig);
    bal0[r] = ballot32(p0);

    float ih1 = fminf(iy2, jy2b) - fmaxf(iy1, jy1b);
    float iw1 = fminf(ix2, jx2b) - fmaxf(ix1, jx1b);
    float in1 = fmaxf(ih1, 0.0f) * fmaxf(iw1, 0.0f);
    float un1 = ub1[r] - in1;
    bool  p1  = (in1 / fmaxf(un1, 1e-9f) > IOU_T) && (jg1 > ig);
    bal1[r] = ballot32(p1);
  }

  if (lane == 0) {
    int jword = blockIdx.x * (JT / 32) + wv;
    #pragma unroll
    for (int r = 0; r < 8; ++r) {
      sup[(i0 + r)     * KWORDS + jword] = (bal0[r] & 0xFFFFu) | (bal1[r] << 16);
      sup[(i0 + 8 + r) * KWORDS + jword] = (bal0[r] >> 16) | (bal1[r] & 0xFFFF0000u);
    }
  }
}

// ---- kernel 5: blocked greedy scan (32 rows per step, 1 workgroup) -------
// Within a 32-row block, aliveness depends only on the block's "diagonal"
// word (row i clears only bits j>i), so one thread resolves it serially;
// then keep_final = keep & ~OR(sup rows of alive i) applied in parallel.

__global__ void __launch_bounds__(KWORDS) scan_kernel(const float* __restrict__ top_s,
                                                      const unsigned* __restrict__ sup,
                                                      unsigned* __restrict__ keepw) {
  __shared__ unsigned keepsh[KWORDS];
  __shared__ unsigned diagrow[32];
  __shared__ unsigned aliveMask;
  int t = threadIdx.x;  // each thread owns one 32-bit keep word
  unsigned w = 0;
  for (int l = 0; l < 32; ++l)
    if (top_s[t * 32 + l] > -0.5f) w |= (1u << l);
  keepsh[t] = w;
  __syncthreads();

  for (int b = 0; b < KWORDS; ++b) {
    if (t < 32) {
      async_copy_b32(&diagrow[t], &sup[(size_t)(32 * b + t) * KWORDS + b]);
      wait_async0();
    }
    __syncthreads();
    if (t == 0) {
      unsigned wb = keepsh[b];
      unsigned alive = 0;
      #pragma unroll 4
      for (int l = 0; l < 32; ++l) {
        if ((wb >> l) & 1u) { alive |= (1u << l); wb &= ~diagrow[l]; }
      }
      aliveMask = alive;
    }
    __syncthreads();
    unsigned am = aliveMask;
    unsigned acc = 0;
    while (am) {
      int l = __builtin_ctz(am);
      am &= am - 1;
      acc |= sup[(size_t)(32 * b + l) * KWORDS + t];
    }
    keepsh[t] &= ~acc;
    __syncthreads();
  }
  keepw[t] = keepsh[t];
}

// ---- kernel 6: second top-k == parallel compaction -----------------------
// Kept entries in ascending j (= descending score), then non-kept ascending:
// exactly jax.lax.top_k order over {kept scores} U {-1 ties}.

__global__ void __launch_bounds__(KWORDS) compact_kernel(const unsigned* __restrict__ keepw,
                                                         const float* __restrict__ top_s,
                                                         int* __restrict__ sel,
                                                         float* __restrict__ out_s) {
  __shared__ unsigned wds[KWORDS];
  __shared__ int pk[KWORDS + 1], pn[KWORDS];
  int t = threadIdx.x;
  unsigned w = keepw[t];
  wds[t] = w;
  __syncthreads();
  if (t == 0) {
    int a = 0, bnk = 0;
    for (int q = 0; q < KWORDS; ++q) {
      pk[q] = a; pn[q] = bnk;
      int c = __popc(wds[q]);
      a += c; bnk += 32 - c;
    }
    pk[KWORDS] = a;  // total kept
  }
  __syncthreads();
  int totalKeep = pk[KWORDS];

  int p = pk[t];
  unsigned wk = w;
  while (wk) {
    int l = __builtin_ctz(wk); wk &= wk - 1;
    if (p < MAX_DET) { int j = t * 32 + l; sel[p] = j; out_s[p] = top_s[j]; }
    ++p;
  }
  p = totalKeep + pn[t];
  unsigned wn = ~w;
  while (wn) {
    int l = __builtin_ctz(wn); wn &= wn - 1;
    if (p < MAX_DET) { int j = t * 32 + l; sel[p] = j; out_s[p] = -1.0f; }
    ++p;
  }
}

// ---- kernel 7: gather final outputs (boxes/classes/masks) ----------------

__global__ void gather_kernel(const float* __restrict__ x,
                              const float* __restrict__ yxyx,
                              const float* __restrict__ classes,
                              const int* __restrict__ top_i,
                              const int* __restrict__ sel,
                              float* __restrict__ out_boxes,
                              float* __restrict__ out_classes,
                              float* __restrict__ out_masks) {
  int d = blockIdx.x;   // 0..299
  int l = threadIdx.x;  // 0..31
  int j = sel[d];
  int gi = top_i[j];
  out_masks[d * NMASK + l] = x[(long long)gi * DIMS + 5 + NCLS + l];
  if (l < 4)  out_boxes[d * 4 + l] = yxyx[gi * 4 + l];
  if (l == 4) out_classes[d] = classes[gi];
}

// ---- launch --------------------------------------------------------------

extern "C" void kernel_launch(void* const* d_in, const int* in_sizes, int n_in,
                              void* d_out, int out_size, void* d_ws, size_t ws_size,
                              hipStream_t stream) {
  (void)in_sizes; (void)n_in; (void)out_size; (void)ws_size;
  const float* x = (const float*)d_in[0];
  float* out = (float*)d_out;  // [0:1200) boxes | [1200:1500) classes | [1500:1800) scores | [1800:11400) masks

  char* ws = (char*)d_ws;
  size_t off = 0;
  auto alloc = [&](size_t bytes) -> char* {
    char* p = ws + off;
    off += (bytes + 255) & ~(size_t)255;
    return p;
  };
  unsigned long long* keys = (unsigned long long*)alloc((size_t)NSORT * 8);
  float* scores_m = (float*)alloc((size_t)N_ANCHORS * 4);
  float* classes  = (float*)alloc((size_t)N_ANCHORS * 4);
  float* yxyx     = (float*)alloc((size_t)N_ANCHORS * 4 * 4);
  float* top_s    = (float*)alloc((size_t)TOPK * 4);
  int*   top_i    = (int*)alloc((size_t)TOPK * 4);
  float* by1      = (float*)alloc((size_t)TOPK * 4);
  float* bx1      = (float*)alloc((size_t)TOPK * 4);
  float* by2      = (float*)alloc((size_t)TOPK * 4);
  float* bx2      = (float*)alloc((size_t)TOPK * 4);
  unsigned* sup   = (unsigned*)alloc((size_t)TOPK * KWORDS * 4);  // 2 MB
  unsigned* keepw = (unsigned*)alloc((size_t)KWORDS * 4);
  int* sel        = (int*)alloc((size_t)MAX_DET * 4);

  prep_kernel<<<(NSORT + 255) / 256, 256, 0, stream>>>(x, scores_m, classes, yxyx, keys);

  // Sort: local full sort of 1024-chunks, then merge phases with global
  // steps only for j >= 1024 and an LDS kernel for j = 512..1.
  bitonic_local_full<<<NSORT / CHUNK, SORT_BLK, 0, stream>>>(keys);
  for (int k = CHUNK * 2; k <= NSORT; k <<= 1) {
    for (int j = k >> 1; j >= CHUNK; j >>= 1)
      bitonic_step<<<NSORT / 256, 256, 0, stream>>>(keys, j, k);
    bitonic_local_merge<<<NSORT / CHUNK, SORT_BLK, 0, stream>>>(keys, k);
  }

  extract_kernel<<<TOPK / 256, 256, 0, stream>>>(keys, scores_m, yxyx,
                                                 top_s, top_i, by1, bx1, by2, bx2);

  dim3 supgrid(TOPK / JT, TOPK / ITILE);   // (16, 256)
  sup_kernel<<<supgrid, JT, 0, stream>>>(by1, bx1, by2, bx2, sup);

  scan_kernel<<<1, KWORDS, 0, stream>>>(top_s, sup, keepw);

  compact_kernel<<<1, KWORDS, 0, stream>>>(keepw, top_s, sel, out + 1500);

  gather_kernel<<<MAX_DET, 32, 0, stream>>>(x, yxyx, classes, top_i, sel,
                                            out + 0, out + 1200, out + 1800);
}